// KCompetitive_7730941133274
// MI455X (gfx1250) — compile-verified
//
#include <hip/hip_runtime.h>
#include <cstdint>

// ---------------- problem constants ----------------
constexpr int KDIM = 2048;     // row length
constexpr int K1 = 64;         // top-k positives
constexpr int K2 = 64;         // top-k negative magnitudes
constexpr float KFACTOR = 6.26f;
constexpr int NTHREADS = 256;  // 8 wave32 waves per row
constexpr int EPT = KDIM / NTHREADS;  // 8 elements per thread (contiguous 32B)

// ---------------- CDNA5 async global->LDS helpers ----------------
typedef int v4i_t __attribute__((vector_size(4 * sizeof(int))));
typedef __attribute__((address_space(1))) v4i_t* gptr_v4i;
typedef __attribute__((address_space(3))) v4i_t* lptr_v4i;

#if defined(__AMDGCN__) && __has_builtin(__builtin_amdgcn_global_load_async_to_lds_b128)
#define USE_ASYNC_BUILTIN 1
#endif

__device__ __forceinline__ void async_copy_b128(const void* gsrc, unsigned lds_off) {
#ifdef USE_ASYNC_BUILTIN
  __builtin_amdgcn_global_load_async_to_lds_b128(
      (gptr_v4i)(uintptr_t)gsrc,
      (lptr_v4i)(uintptr_t)lds_off,
      0, 0);
#else
  asm volatile("global_load_async_to_lds_b128 %0, %1, off"
               :: "v"(lds_off), "v"(gsrc) : "memory");
#endif
}

__device__ __forceinline__ void wait_asynccnt0() {
#if defined(__AMDGCN__) && __has_builtin(__builtin_amdgcn_s_wait_asynccnt)
  __builtin_amdgcn_s_wait_asynccnt(0);
#else
  asm volatile("s_wait_asynccnt 0" ::: "memory");
#endif
}

// ---------------- wave32 reduction ----------------
__device__ __forceinline__ float wave_sum(float v) {
#pragma unroll
  for (int m = 16; m >= 1; m >>= 1) v += __shfl_xor(v, m, 32);
  return v;
}

// ---------------- kernel: one workgroup per row ----------------
__global__ __launch_bounds__(NTHREADS)
void kcomp_tanh_kernel(const float* __restrict__ x, float* __restrict__ out) {
  __shared__ float s_row[KDIM];        // 8 KB staged row
  __shared__ unsigned s_histP[256];    // per-round radix histogram (positives)
  __shared__ unsigned s_histN[256];    // per-round radix histogram (neg magnitudes)
  __shared__ unsigned s_wsum[16];      // 8 wave totals for P, 8 for N
  __shared__ unsigned s_pick[4];       // binP, subP, binN, subN
  __shared__ unsigned s_claim[2];
  __shared__ float s_f[4];             // sumP_all, sumP_gt, sumN_all, sumN_gt

  const unsigned t = threadIdx.x;
  const unsigned lane = t & 31u;
  const unsigned wave = t >> 5;
  const size_t rowBase = (size_t)blockIdx.x * KDIM;

  // ---- stage the row into LDS with async b128 copies (2 x 16B per lane = 8 KB) ----
  {
    const char* g = (const char*)(x + rowBase) + (size_t)t * 16u;
    const unsigned l = ((unsigned)(uintptr_t)&s_row[0]) + t * 16u;
    async_copy_b128(g, l);
    async_copy_b128(g + 4096, l + 4096);
  }
  s_histP[t] = 0u;
  s_histN[t] = 0u;
  if (t < 4) s_f[t] = 0.0f;
  if (t < 2) s_claim[t] = 0u;
  wait_asynccnt0();
  __syncthreads();

  // ---- pull this thread's contiguous 32B into registers via 2x ds_load_b128 ----
  float xv[EPT];
  unsigned up[EPT], un[EPT];
  {
    const float4* s_row4 = reinterpret_cast<const float4*>(s_row);
    const float4 a = s_row4[t * 2u];
    const float4 b = s_row4[t * 2u + 1u];
    xv[0] = a.x; xv[1] = a.y; xv[2] = a.z; xv[3] = a.w;
    xv[4] = b.x; xv[5] = b.y; xv[6] = b.z; xv[7] = b.w;
  }
#pragma unroll
  for (int i = 0; i < EPT; ++i) {
    const unsigned b = __float_as_uint(xv[i]);
    up[i] = (xv[i] > 0.0f) ? b : 0u;                  // positive float bits are monotone
    un[i] = (xv[i] < 0.0f) ? (b & 0x7FFFFFFFu) : 0u;  // |x| for negatives
  }

  // ---- fused dual radix select (4 rounds, 3 barriers/round) ----
  unsigned prefP = 0u, prefN = 0u;
  int rankP = K1, rankN = K2;
  for (int shift = 24; shift >= 0; shift -= 8) {
    const unsigned himask = (shift == 24) ? 0u : (0xFFFFFFFFu << (shift + 8));
#pragma unroll
    for (int i = 0; i < EPT; ++i) {
      if ((up[i] & himask) == prefP) atomicAdd(&s_histP[(up[i] >> shift) & 255u], 1u);
      if ((un[i] & himask) == prefN) atomicAdd(&s_histN[(un[i] >> shift) & 255u], 1u);
    }
    __syncthreads();
    const unsigned hP = s_histP[t];
    const unsigned hN = s_histN[t];
    s_histP[t] = 0u;  // re-arm for next round; next atomics are 2 barriers away
    s_histN[t] = 0u;
    // intra-wave suffix scan over this wave's 32 bins (no barriers)
    unsigned geP = hP, geN = hN;
#pragma unroll
    for (int off = 1; off < 32; off <<= 1) {
      const unsigned aP = __shfl_down(geP, off, 32);
      const unsigned aN = __shfl_down(geN, off, 32);
      if (lane + (unsigned)off < 32u) { geP += aP; geN += aN; }
    }
    if (lane == 0u) {
      s_wsum[wave] = geP;
      s_wsum[8u + wave] = geN;
    }
    __syncthreads();
    // add totals of higher waves (higher bins)
#pragma unroll
    for (int w = 0; w < 8; ++w) {
      if (w > (int)wave) {
        geP += s_wsum[w];
        geN += s_wsum[8 + w];
      }
    }
    const unsigned geP1 = geP - hP;  // suffix count starting at bin t+1
    const unsigned geN1 = geN - hN;
    if ((int)geP >= rankP && (int)geP1 < rankP) { s_pick[0] = t; s_pick[1] = geP1; }
    if ((int)geN >= rankN && (int)geN1 < rankN) { s_pick[2] = t; s_pick[3] = geN1; }
    __syncthreads();
    prefP |= s_pick[0] << shift;
    rankP -= (int)s_pick[1];
    prefN |= s_pick[2] << shift;
    rankN -= (int)s_pick[3];
  }
  const unsigned Tp = prefP;  // value of k-th largest positive (bits)
  const unsigned Tn = prefN;  // value of k-th largest |negative| (bits)

  // ---- sums + tie claiming in one pass ----
  float sPa = 0.f, sPg = 0.f, sNa = 0.f, sNg = 0.f;
  unsigned pw = 0u, nw = 0u;
#pragma unroll
  for (int i = 0; i < EPT; ++i) {
    const float xi = xv[i];
    const float p = (xi > 0.0f) ? xi : 0.0f;
    const float n = (xi < 0.0f) ? -xi : 0.0f;
    sPa += p;
    sNa += n;
    if (up[i] > Tp) {
      sPg += p;
      pw |= 1u << i;
    } else if (up[i] == Tp) {
      if (atomicAdd(&s_claim[0], 1u) < (unsigned)rankP) pw |= 1u << i;
    }
    if (un[i] > Tn) {
      sNg += n;
      nw |= 1u << i;
    } else if (un[i] == Tn) {
      if (atomicAdd(&s_claim[1], 1u) < (unsigned)rankN) nw |= 1u << i;
    }
  }
  sPa = wave_sum(sPa);
  sPg = wave_sum(sPg);
  sNa = wave_sum(sNa);
  sNg = wave_sum(sNg);
  if (lane == 0u) {
    atomicAdd(&s_f[0], sPa);
    atomicAdd(&s_f[1], sPg);
    atomicAdd(&s_f[2], sNa);
    atomicAdd(&s_f[3], sNg);
  }
  __syncthreads();

  // winners at the threshold contribute exactly rank * T
  const float Tpf = __uint_as_float(Tp);
  const float Tnf = __uint_as_float(Tn);
  const float Ptmp = KFACTOR * (s_f[0] - s_f[1] - (float)rankP * Tpf);
  const float Ntmp = KFACTOR * (s_f[2] - s_f[3] - (float)rankN * Tnf);

  // ---- out = P_reset2 - N_reset2, written as 2x global_store_b128 ----
  float o[EPT];
#pragma unroll
  for (int i = 0; i < EPT; ++i) {
    float v = 0.0f;
    if (pw & (1u << i))
      v = xv[i] + Ptmp;
    else if (nw & (1u << i))
      v = xv[i] - Ntmp;
    o[i] = v;
  }
  {
    float4* out4 = reinterpret_cast<float4*>(out + rowBase);
    float4 o0, o1;
    o0.x = o[0]; o0.y = o[1]; o0.z = o[2]; o0.w = o[3];
    o1.x = o[4]; o1.y = o[5]; o1.z = o[6]; o1.w = o[7];
    out4[t * 2u] = o0;
    out4[t * 2u + 1u] = o1;
  }
}

extern "C" void kernel_launch(void* const* d_in, const int* in_sizes, int n_in,
                              void* d_out, int out_size, void* d_ws, size_t ws_size,
                              hipStream_t stream) {
  const float* x = (const float*)d_in[0];
  float* out = (float*)d_out;
  const int rows = in_sizes[0] / KDIM;  // 16384
  kcomp_tanh_kernel<<<dim3(rows), dim3(NTHREADS), 0, stream>>>(x, out);
}